// Model_67491116089495
// MI455X (gfx1250) — compile-verified
//
#include <hip/hip_runtime.h>
#include <hip/hip_bf16.h>
#include <stdint.h>

// ---------------- problem constants (from reference) ----------------
#define BATCH 32
#define CCH   3
#define HH    192
#define WW    640
#define HW    (HH * WW)          // 122880
#define NV    8
#define TOPK  30
#define DOWN_SAMPLE 4.0f
#define SCORE_THRESH 0.3f

// ---------------- tiling ----------------
#define TR    12                 // rows per block tile
#define RB    (HH / TR)          // 16 row-blocks per (b,c)
#define BLOCKS_PER_BATCH (CCH * RB)   // 48
#define NCAND (BLOCKS_PER_BATCH * TOPK) // 1440 merged candidates per batch
#define CAP   2048               // per-block LDS candidate capacity
#define NT    256                // 8 wave32s

typedef __attribute__((ext_vector_type(4))) unsigned int u32x4;
typedef __attribute__((ext_vector_type(8))) int          i32x8;
typedef __attribute__((ext_vector_type(4))) int          i32x4;

__device__ __forceinline__ float sigmoidf_fast(float x) {
  return 1.0f / (1.0f + __expf(-x));
}

// monotonic float -> uint32 key (greater key == greater float)
__device__ __forceinline__ unsigned ord_f32(float f) {
  unsigned u = __float_as_uint(f);
  return (u & 0x80000000u) ? ~u : (u | 0x80000000u);
}
__device__ __forceinline__ float unord_f32(unsigned ou) {
  unsigned u = (ou & 0x80000000u) ? (ou ^ 0x80000000u) : ~ou;
  return __uint_as_float(u);
}

// =====================================================================
// Kernel 1: TDM-staged tile -> NMS local-max test on raw logits ->
//           per-block exact top-30 keys into workspace.
// key = ord(logit) << 32 | ~flatIdx  (desc value, tie -> lower index)
// =====================================================================
__global__ __launch_bounds__(NT)
void nms_block_topk_kernel(const float* __restrict__ logits,
                           unsigned long long* __restrict__ ws) {
  const int tid  = threadIdx.x;
  const int rblk = blockIdx.x % RB;
  const int c    = (blockIdx.x / RB) % CCH;
  const int b    = blockIdx.x / (RB * CCH);
  const int r0   = rblk * TR;

  __shared__ float              tile[(TR + 2) * WW];   // 35.8 KB
  __shared__ unsigned long long cand[CAP];             // 16 KB
  __shared__ unsigned long long skey[NT];
  __shared__ int                sslot[NT];
  __shared__ unsigned long long winners[TOPK];
  __shared__ int                cnt;

  if (tid == 0) cnt = 0;
  const float NEG = -__builtin_inff();
  // -inf halo rows at image top/bottom (matches reduce_window -inf pad)
  if (r0 == 0)       for (int x = tid; x < WW; x += NT) tile[x] = NEG;
  if (r0 + TR == HH) for (int x = tid; x < WW; x += NT) tile[(TR + 1) * WW + x] = NEG;

  // ---- TDM: DMA rows [g0, g0+nrows) of this (b,c) image into LDS ----
  int g0 = r0 - 1;            if (g0 < 0) g0 = 0;
  int g1 = r0 + TR + 1;       if (g1 > HH) g1 = HH;
  const int nrows = g1 - g0;
  const int lr0   = g0 - (r0 - 1);        // 0 interior, 1 at image top

  if (tid < 32) {  // one wave issues the descriptor (TDM ignores EXEC)
    unsigned long long gaddr =
        (unsigned long long)(uintptr_t)(logits + (((size_t)b * CCH + c) * HH + g0) * WW);
    unsigned lds_addr = (unsigned)(uintptr_t)(&tile[lr0 * WW]);

    u32x4 d0;   // D# group 0: count=1, lds_addr, global_addr, type=2
    d0[0] = 1u;
    d0[1] = lds_addr;
    d0[2] = (unsigned)(gaddr & 0xFFFFFFFFull);
    d0[3] = (unsigned)((gaddr >> 32) & 0x01FFFFFFull) | 0x80000000u; // type=2 -> bits[127:126]=10b

    i32x8 d1;   // D# group 1: data_size=4B, dims/strides for a 2-D tile
    d1[0] = (int)(2u << 16);                 // workgroup_mask=0, data_size=2 (4 bytes)
    d1[1] = (int)((WW & 0xFFFF) << 16);      // tensor_dim0[15:0] in bits[63:48]
    d1[2] = (int)((nrows & 0xFFFF) << 16);   // tensor_dim0[31:16]=0 | tensor_dim1[15:0]
    d1[3] = (int)((WW & 0xFFFF) << 16);      // tensor_dim1[31:16]=0 | tile_dim0
    d1[4] = (int)(nrows & 0xFFFF);           // tile_dim1 | tile_dim2=0
    d1[5] = WW;                              // tensor_dim0_stride[31:0]
    d1[6] = 0;                               // stride hi, tensor_dim1_stride lo
    d1[7] = 0;

    i32x4 z4; z4[0] = z4[1] = z4[2] = z4[3] = 0;  // groups 2/3 unused (2-D tile)
    i32x8 z8; z8[0] = z8[1] = z8[2] = z8[3] = z8[4] = z8[5] = z8[6] = z8[7] = 0;

    // 6-arg form on this toolchain (clang-23 / therock-10.0 headers)
    __builtin_amdgcn_tensor_load_to_lds(d0, d1, z4, z4, z8, 0);
    __builtin_amdgcn_s_wait_tensorcnt(0);
  }
  __syncthreads();

  // ---- local-max test (monotonic: compare raw logits, no sigmoid) ----
  const int base = c * HW;
  for (int p = tid; p < TR * WW; p += NT) {
    const int lx   = p % WW;
    const int lrow = p / WW;        // 0..TR-1
    const int lr   = lrow + 1;      // LDS row
    const float* rm = &tile[(lr - 1) * WW + lx];
    const float* rc = &tile[lr * WW + lx];
    const float* rp = &tile[(lr + 1) * WW + lx];
    const float v = rc[0];
    float m = fmaxf(rm[0], rp[0]);
    if (lx > 0)      m = fmaxf(m, fmaxf(rm[-1], fmaxf(rc[-1], rp[-1])));
    if (lx < WW - 1) m = fmaxf(m, fmaxf(rm[1],  fmaxf(rc[1],  rp[1])));
    if (v >= m) {   // v == max of 3x3 window
      unsigned flat = (unsigned)(base + (r0 + lrow) * WW + lx);
      unsigned long long key =
          ((unsigned long long)ord_f32(v) << 32) | (unsigned)(~flat);
      int slot = atomicAdd(&cnt, 1);
      if (slot < CAP) cand[slot] = key;
    }
  }
  __syncthreads();
  int n = cnt; if (n > CAP) n = CAP;

  // ---- exact block top-30 by iterative argmax + claim ----
  for (int r = 0; r < TOPK; ++r) {
    unsigned long long bk = 0ull; int bs = -1;
    for (int i = tid; i < n; i += NT) {
      unsigned long long k = cand[i];
      if (k > bk) { bk = k; bs = i; }
    }
    skey[tid] = bk; sslot[tid] = bs;
    __syncthreads();
    for (int s = NT / 2; s > 0; s >>= 1) {
      if (tid < s && skey[tid + s] > skey[tid]) {
        skey[tid] = skey[tid + s]; sslot[tid] = sslot[tid + s];
      }
      __syncthreads();
    }
    if (tid == 0) {
      winners[r] = skey[0];
      if (sslot[0] >= 0) cand[sslot[0]] = 0ull;   // claim
    }
    __syncthreads();
  }

  if (tid < TOPK)
    ws[((size_t)b * BLOCKS_PER_BATCH + (c * RB + rblk)) * TOPK + tid] = winners[tid];
}

// =====================================================================
// Kernel 2: per-batch merge of 1440 keys -> exact sorted top-30, then
//           gathers + projection math + concatenated float outputs.
// =====================================================================
__global__ __launch_bounds__(NT)
void merge_gather_kernel(const unsigned long long* __restrict__ ws,
                         const float* __restrict__ off,     // [B,16,H,W]
                         const float* __restrict__ moff,    // [B,2,H,W]
                         float* __restrict__ out) {
  const int b = blockIdx.x, tid = threadIdx.x;

  __shared__ unsigned long long keys[NCAND];
  __shared__ unsigned long long skey[NT];
  __shared__ int                sslot[NT];
  __shared__ unsigned long long winners[TOPK];

  for (int i = tid; i < NCAND; i += NT) keys[i] = ws[(size_t)b * NCAND + i];
  __syncthreads();

  for (int r = 0; r < TOPK; ++r) {
    unsigned long long bk = 0ull; int bs = -1;
    for (int i = tid; i < NCAND; i += NT) {
      unsigned long long k = keys[i];
      if (k > bk) { bk = k; bs = i; }
    }
    skey[tid] = bk; sslot[tid] = bs;
    __syncthreads();
    for (int s = NT / 2; s > 0; s >>= 1) {
      if (tid < s && skey[tid + s] > skey[tid]) {
        skey[tid] = skey[tid + s]; sslot[tid] = sslot[tid + s];
      }
      __syncthreads();
    }
    if (tid == 0) {
      winners[r] = skey[0];
      if (sslot[0] >= 0) keys[sslot[0]] = 0ull;
    }
    __syncthreads();
  }

  // ---- per-point decode + gather + math (30 lanes of wave 0) ----
  if (tid < TOPK) {
    const int r = tid;
    const unsigned long long key = winners[r];

    // output section offsets (floats), concatenated in return order
    float* o_cls   = out;                                 // [B,30]
    float* o_scr   = out + BATCH * TOPK;                  // 960
    float* o_mprj  = out + 2 * BATCH * TOPK;              // 1920
    float* o_vprj  = out + 4 * BATCH * TOPK;              // 3840
    float* o_bbox  = out + 4 * BATCH * TOPK + BATCH * TOPK * NV * 2;  // 19200
    float* o_mask  = o_bbox + BATCH * TOPK * 4;           // 23040

    if (key == 0ull) {   // padding (fewer than 30 local maxima)
      o_cls[b * TOPK + r] = 0.0f;
      o_scr[b * TOPK + r] = 0.0f;
      o_mask[b * TOPK + r] = 0.0f;
      o_mprj[(b * TOPK + r) * 2 + 0] = 0.0f;
      o_mprj[(b * TOPK + r) * 2 + 1] = 0.0f;
      for (int v = 0; v < NV; ++v) {
        o_vprj[((b * TOPK + r) * NV + v) * 2 + 0] = 0.0f;
        o_vprj[((b * TOPK + r) * NV + v) * 2 + 1] = 0.0f;
      }
      for (int j = 0; j < 4; ++j) o_bbox[(b * TOPK + r) * 4 + j] = 0.0f;
    } else {
      const unsigned flat = ~(unsigned)(key & 0xFFFFFFFFull);
      const float logit   = unord_f32((unsigned)(key >> 32));
      const float score   = sigmoidf_fast(logit);

      const int cls = (int)(flat / (unsigned)HW);
      const unsigned rem = flat % (unsigned)HW;
      const int y = (int)(rem / (unsigned)WW);
      const int x = (int)(rem % (unsigned)WW);

      const size_t pix = (size_t)y * WW + x;
      const float s0 = sigmoidf_fast(moff[((size_t)b * 2 + 0) * HW + pix]);
      const float s1 = sigmoidf_fast(moff[((size_t)b * 2 + 1) * HW + pix]);
      const float xf = (float)x + s0;
      const float yf = (float)y + s1;

      o_cls[b * TOPK + r]  = (float)cls;
      o_scr[b * TOPK + r]  = score;
      o_mask[b * TOPK + r] = (score > SCORE_THRESH) ? 1.0f : 0.0f;
      o_mprj[(b * TOPK + r) * 2 + 0] = DOWN_SAMPLE * xf;
      o_mprj[(b * TOPK + r) * 2 + 1] = DOWN_SAMPLE * yf;

      float mnx =  3.4e38f, mny =  3.4e38f, mxx = -3.4e38f, mxy = -3.4e38f;
      for (int v = 0; v < NV; ++v) {
        const float ox = off[((size_t)b * (2 * NV) + 2 * v + 0) * HW + pix];
        const float oy = off[((size_t)b * (2 * NV) + 2 * v + 1) * HW + pix];
        const float px = DOWN_SAMPLE * (ox + xf);
        const float py = DOWN_SAMPLE * (oy + yf);
        o_vprj[((b * TOPK + r) * NV + v) * 2 + 0] = px;
        o_vprj[((b * TOPK + r) * NV + v) * 2 + 1] = py;
        mnx = fminf(mnx, px); mny = fminf(mny, py);
        mxx = fmaxf(mxx, px); mxy = fmaxf(mxy, py);
      }
      o_bbox[(b * TOPK + r) * 4 + 0] = mnx;
      o_bbox[(b * TOPK + r) * 4 + 1] = mny;
      o_bbox[(b * TOPK + r) * 4 + 2] = mxx;
      o_bbox[(b * TOPK + r) * 4 + 3] = mxy;
    }
  }
}

extern "C" void kernel_launch(void* const* d_in, const int* in_sizes, int n_in,
                              void* d_out, int out_size, void* d_ws, size_t ws_size,
                              hipStream_t stream) {
  (void)in_sizes; (void)n_in; (void)out_size; (void)ws_size;
  const float* main_kf = (const float*)d_in[0];   // [B,3,H,W]
  const float* off     = (const float*)d_in[1];   // [B,16,H,W]
  const float* moff    = (const float*)d_in[2];   // [B,2,H,W]
  // d_in[3] (vertex_offset_kf_logits) is unused by the reference outputs.

  unsigned long long* cand_ws = (unsigned long long*)d_ws;  // B*1440 u64 = 360 KB

  nms_block_topk_kernel<<<BATCH * CCH * RB, NT, 0, stream>>>(main_kf, cand_ws);
  merge_gather_kernel<<<BATCH, NT, 0, stream>>>(cand_ws, off, moff, (float*)d_out);
}